// TernaryActivation_75393855914347
// MI455X (gfx1250) — compile-verified
//
#include <hip/hip_runtime.h>
#include <hip/hip_bf16.h>

// ---------------------------------------------------------------------------
// TernaryActivation (BitNet-style), fp32, N = 32*4096*1024 = 134,217,728
// Memory-bound 3-pass pipeline (2 GB total traffic ~ 86us @ 23.3 TB/s):
//   k1: partial sums of |x|             (read 512MB, NT)
//   k2: finalize delta = 0.7*mean(|x|)  (tiny)
//   k3: partial masked sums + counts    (read 512MB, NT)
//   k4: finalize alpha                  (tiny)
//   k5: out = alpha*sign(x)*(|x|>=delta)(read 512MB + write 512MB, NT)
// Working set (512MB) > L2 (192MB) with no inter-pass reuse -> all bulk
// traffic uses non-temporal hints (TH=NT) via __builtin_nontemporal_*.
// All reductions are fixed-order trees (deterministic, no float atomics).
// Wave-level reduction uses V_WMMA_F32_16X16X4_F32 with B = ones
// (layout-proof) per documented A/D VGPR layouts: D[m,n] = p[m] + p[m+16].
// ---------------------------------------------------------------------------

typedef float v2f __attribute__((ext_vector_type(2)));
typedef float v4f __attribute__((ext_vector_type(4)));
typedef float v8f __attribute__((ext_vector_type(8)));

#define NB_RED 2048
#define NT 256

// Full-wave (32 lane) sum. EXEC must be all-ones at every call site
// (all calls are wave-uniform). Exact fp32 RNE accumulation.
__device__ __forceinline__ float wave_reduce_wmma(float v) {
  v2f a; a[0] = v;    a[1] = 0.0f;   // A[16x4]: lane L<16 -> (M=L,K=0|1); L>=16 -> (M=L-16,K=2|3)
  v2f b; b[0] = 1.0f; b[1] = 1.0f;   // B[4x16] = ones (any layout of ones is ones)
  v8f c = {};
  c = __builtin_amdgcn_wmma_f32_16x16x4_f32(false, a, false, b, (short)0, c,
                                            false, false);
  // D[m,n] = rowsum_A(m) = p[m] + p[m+16].
  // Lanes 0-15 hold D[v, lane] = rowsum(v); lanes 16-31 hold rowsum(v+8).
  float t = ((c[0] + c[1]) + (c[2] + c[3])) + ((c[4] + c[5]) + (c[6] + c[7]));
  t += __shfl_xor(t, 16, 32);        // combine rowsum(0..7) and rowsum(8..15)
  return t;                          // every lane: full 32-lane sum
}

__device__ __forceinline__ float block_reduce_sum(float v) {
  __shared__ float sdata[NT / 32];
  const int lane = threadIdx.x & 31;
  const int wid  = threadIdx.x >> 5;
  float t = wave_reduce_wmma(v);
  if (lane == 0) sdata[wid] = t;
  __syncthreads();
  float r = 0.0f;
  if (wid == 0) {
    float w = (lane < (int)(blockDim.x >> 5)) ? sdata[lane] : 0.0f;
    r = wave_reduce_wmma(w);
    if (lane == 0) sdata[0] = r;
  }
  __syncthreads();
  r = sdata[0];
  __syncthreads();
  return r;
}

// ---- pass 1: sum |x| -------------------------------------------------------
__global__ void k_abs_sum(const v4f* __restrict__ x4, long long nvec,
                          const float* __restrict__ xs, long long n,
                          float* __restrict__ p1) {
  v4f acc = {0.f, 0.f, 0.f, 0.f};
  const long long stride = (long long)gridDim.x * blockDim.x;
  for (long long i = (long long)blockIdx.x * blockDim.x + threadIdx.x; i < nvec;
       i += stride) {
    v4f v = __builtin_nontemporal_load(&x4[i]);
    acc[0] += fabsf(v[0]); acc[1] += fabsf(v[1]);
    acc[2] += fabsf(v[2]); acc[3] += fabsf(v[3]);
  }
  float s = (acc[0] + acc[1]) + (acc[2] + acc[3]);
  if (blockIdx.x == 0) {  // scalar tail (N % 4 != 0 safety)
    for (long long i = nvec * 4 + threadIdx.x; i < n; i += blockDim.x)
      s += fabsf(xs[i]);
  }
  float tot = block_reduce_sum(s);
  if (threadIdx.x == 0) p1[blockIdx.x] = tot;
}

// ---- pass 2: delta = 0.7 * mean -------------------------------------------
__global__ void k_finalize_delta(const float* __restrict__ p1, int nb,
                                 long long n, float* __restrict__ scal) {
  float s = 0.0f;
  for (int i = threadIdx.x; i < nb; i += blockDim.x) s += p1[i];
  float tot = block_reduce_sum(s);
  if (threadIdx.x == 0) scal[0] = 0.7f * (tot / (float)n);
}

// ---- pass 3: masked sum + count -------------------------------------------
__global__ void k_masked_sum(const v4f* __restrict__ x4, long long nvec,
                             const float* __restrict__ xs, long long n,
                             const float* __restrict__ scal,
                             float* __restrict__ p2, float* __restrict__ c2) {
  const float delta = scal[0];
  v4f acc = {0.f, 0.f, 0.f, 0.f};
  unsigned cnt = 0u;
  const long long stride = (long long)gridDim.x * blockDim.x;
  for (long long i = (long long)blockIdx.x * blockDim.x + threadIdx.x; i < nvec;
       i += stride) {
    v4f v = __builtin_nontemporal_load(&x4[i]);
    float ax = fabsf(v[0]), ay = fabsf(v[1]), az = fabsf(v[2]), aw = fabsf(v[3]);
    bool mx = ax >= delta, my = ay >= delta, mz = az >= delta, mw = aw >= delta;
    acc[0] += mx ? ax : 0.0f; acc[1] += my ? ay : 0.0f;
    acc[2] += mz ? az : 0.0f; acc[3] += mw ? aw : 0.0f;
    cnt += (unsigned)mx + (unsigned)my + (unsigned)mz + (unsigned)mw;
  }
  float s = (acc[0] + acc[1]) + (acc[2] + acc[3]);
  float c = (float)cnt;  // <= elements/thread, exact in fp32
  if (blockIdx.x == 0) {
    for (long long i = nvec * 4 + threadIdx.x; i < n; i += blockDim.x) {
      float ax = fabsf(xs[i]);
      if (ax >= delta) { s += ax; c += 1.0f; }
    }
  }
  float ts = block_reduce_sum(s);
  float tc = block_reduce_sum(c);  // block count <= 2^17, exact in fp32
  if (threadIdx.x == 0) { p2[blockIdx.x] = ts; c2[blockIdx.x] = tc; }
}

// ---- pass 4: alpha = masked_sum / max(cnt,1) ------------------------------
__global__ void k_finalize_alpha(const float* __restrict__ p2,
                                 const float* __restrict__ c2, int nb,
                                 float* __restrict__ scal) {
  float s = 0.0f, c = 0.0f;
  for (int i = threadIdx.x; i < nb; i += blockDim.x) { s += p2[i]; c += c2[i]; }
  float ts = block_reduce_sum(s);
  float tc = block_reduce_sum(c);
  if (threadIdx.x == 0) scal[1] = ts / fmaxf(tc, 1.0f);
}

// ---- pass 5: out = alpha * trunc(clip(x/delta,-1,1)) ----------------------
__global__ void k_quantize(const v4f* __restrict__ x4, long long nvec,
                           const float* __restrict__ xs, long long n,
                           const float* __restrict__ scal,
                           v4f* __restrict__ o4, float* __restrict__ os) {
  const float delta = scal[0];
  const float alpha = scal[1];  // cnt==0 => alpha==0 => out==0 == q (matches ref)
  const long long stride = (long long)gridDim.x * blockDim.x;
  for (long long i = (long long)blockIdx.x * blockDim.x + threadIdx.x; i < nvec;
       i += stride) {
    v4f v = __builtin_nontemporal_load(&x4[i]);
    v4f o;
    o[0] = (fabsf(v[0]) >= delta) ? copysignf(alpha, v[0]) : 0.0f;
    o[1] = (fabsf(v[1]) >= delta) ? copysignf(alpha, v[1]) : 0.0f;
    o[2] = (fabsf(v[2]) >= delta) ? copysignf(alpha, v[2]) : 0.0f;
    o[3] = (fabsf(v[3]) >= delta) ? copysignf(alpha, v[3]) : 0.0f;
    __builtin_nontemporal_store(o, &o4[i]);
  }
  if (blockIdx.x == 0) {
    for (long long i = nvec * 4 + threadIdx.x; i < n; i += blockDim.x)
      os[i] = (fabsf(xs[i]) >= delta) ? copysignf(alpha, xs[i]) : 0.0f;
  }
}

extern "C" void kernel_launch(void* const* d_in, const int* in_sizes, int n_in,
                              void* d_out, int out_size, void* d_ws,
                              size_t ws_size, hipStream_t stream) {
  const float* x = (const float*)d_in[0];
  float* out = (float*)d_out;
  const long long n = (long long)in_sizes[0];
  const long long nvec = n >> 2;

  // Workspace layout (floats): [0,NB) p1 | [NB,2NB) p2 | [2NB,3NB) c2 | [3NB] delta,alpha
  float* wsf  = (float*)d_ws;
  float* p1   = wsf;
  float* p2   = wsf + NB_RED;
  float* c2   = wsf + 2 * NB_RED;
  float* scal = wsf + 3 * NB_RED;

  const v4f* x4 = (const v4f*)x;
  v4f* o4 = (v4f*)out;

  k_abs_sum<<<NB_RED, NT, 0, stream>>>(x4, nvec, x, n, p1);
  k_finalize_delta<<<1, NT, 0, stream>>>(p1, NB_RED, n, scal);
  k_masked_sum<<<NB_RED, NT, 0, stream>>>(x4, nvec, x, n, scal, p2, c2);
  k_finalize_alpha<<<1, NT, 0, stream>>>(p2, c2, NB_RED, scal);
  k_quantize<<<NB_RED, NT, 0, stream>>>(x4, nvec, x, n, scal, o4, out);
}